// RuntimeCGAAlgebra_74483322847653
// MI455X (gfx1250) — compile-verified
//
#include <hip/hip_runtime.h>

typedef __attribute__((ext_vector_type(2))) float v2f;
typedef __attribute__((ext_vector_type(8))) float v8f;

#define NB_BLADES 256
#define NODD      128

#ifndef __has_builtin
#define __has_builtin(x) 0
#endif

#if __has_builtin(__builtin_amdgcn_global_load_async_to_lds_b32)
#define USE_ASYNC_LDS 1
#else
#define USE_ASYNC_LDS 0
#endif

#if __has_builtin(__builtin_amdgcn_wmma_f32_16x16x4_f32)
#define USE_WMMA_F32 1
#else
#define USE_WMMA_F32 0
#endif

typedef __attribute__((address_space(1))) int* gint_p;
typedef __attribute__((address_space(3))) int* lint_p;

// ---------------------------------------------------------------------------
// ws layout (ints): [0,1024) stage1[128][8], [1024,2048) stage2[128][8]
// entry: bit14 = valid, bit15 = negative sign, bits[7:0] = source slot
// stage1[oddSlot][pointSlot] -> motor slot k  (mx[o] += s * motor[k] * point[p])
// stage2[oddSlot][outSlot]   -> motor slot j  (out[r] += s * mx[i] * mrev[j])
// ---------------------------------------------------------------------------

__global__ void cga_setup(const int* __restrict__ left, const int* __restrict__ right,
                          const int* __restrict__ res,  const int* __restrict__ sgn,
                          const int* __restrict__ pmask, const int* __restrict__ mmask,
                          int nnz, int nmotor, int npoint,
                          int* __restrict__ stage1, int* __restrict__ stage2)
{
    __shared__ int minv[NB_BLADES], pinv[NB_BLADES], oinv[NB_BLADES];
    const int t = threadIdx.x;                 // 256 threads, 1 block
    minv[t] = -1; pinv[t] = -1; oinv[t] = -1;
    __syncthreads();
    if (t < nmotor) minv[mmask[t]] = t;
    if (t < npoint) pinv[pmask[t]] = t;
    __syncthreads();
    if (t == 0) {                              // odd blades = not motor, not pseudoscalar
        int c = 0;                             // pseudoscalar sorts last: index NB-1
        for (int i = 0; i < NB_BLADES; ++i)
            if (minv[i] < 0 && i != NB_BLADES - 1) oinv[i] = c++;
    }
    for (int i = t; i < 2 * NODD * 8; i += blockDim.x) {
        if (i < NODD * 8) stage1[i] = 0;
        else              stage2[i - NODD * 8] = 0;
    }
    __syncthreads();
    for (int e = t; e < nnz; e += blockDim.x) {
        const int l = left[e], r = right[e], o = res[e];
        const int pk = 0x4000 | (sgn[e] < 0 ? 0x8000 : 0);
        // first gp: motor (left) x point (right) -> odd blade (res)
        const int ms = minv[l], ps = pinv[r], os = oinv[o];
        if (ms >= 0 && ps >= 0 && os >= 0) stage1[os * 8 + ps] = pk | ms;
        // second gp: mx/odd (left) x motor_rev (right) -> point blade (res)
        const int oi = oinv[l], mj = minv[r], pr = pinv[o];
        if (oi >= 0 && mj >= 0 && pr >= 0) stage2[oi * 8 + pr] = pk | mj;
    }
}

// ---------------------------------------------------------------------------
// One 128-thread workgroup (4 waves) per batch element.
// ---------------------------------------------------------------------------
__global__ void __launch_bounds__(128)
cga_apply(const float* __restrict__ motor, const float* __restrict__ x,
          const int* __restrict__ mrs,
          const int* __restrict__ stage1, const int* __restrict__ stage2,
          float* __restrict__ out, int nmotor)
{
    const int b = blockIdx.x;
    const int t = threadIdx.x;                  // 0..127

    __shared__ float sm_motor[NODD];
    __shared__ float sm_mrev[NODD];
    __shared__ float sm_point[8];
    __shared__ float sm_mx[NODD];
    __shared__ float sm_part[64];               // 4 rows x 16 cols, zero padded

    if (t < 64) sm_part[t] = 0.0f;
    if (t >= nmotor && t < NODD) { sm_motor[t] = 0.0f; sm_mrev[t] = 0.0f; }

    __builtin_prefetch(stage1 + t * 8, 0, 0);   // global_prefetch_b8
    __builtin_prefetch(stage2 + t * 8, 0, 0);

#if USE_ASYNC_LDS
    if (t < nmotor)
        __builtin_amdgcn_global_load_async_to_lds_b32(
            (gint_p)(motor + (size_t)b * nmotor + t),
            (lint_p)&sm_motor[t], 0, 0);
    if (t < 6)
        __builtin_amdgcn_global_load_async_to_lds_b32(
            (gint_p)(x + (size_t)b * 6 + t),
            (lint_p)&sm_point[t], 0, 0);
#if __has_builtin(__builtin_amdgcn_s_wait_asynccnt)
    __builtin_amdgcn_s_wait_asynccnt(0);
#else
    asm volatile("s_wait_asynccnt 0" ::: "memory");
#endif
#else
    if (t < nmotor) sm_motor[t] = motor[(size_t)b * nmotor + t];
    if (t < 6)      sm_point[t] = x[(size_t)b * 6 + t];
#endif
    if (t < nmotor) sm_mrev[t] = sm_motor[t] * (float)mrs[t];
    __syncthreads();

    if (t == 0) {                               // conformal embedding tail
        float h = 0.0f;
        #pragma unroll
        for (int p = 0; p < 6; ++p) h += sm_point[p] * sm_point[p];
        h *= 0.5f;
        sm_point[6] = h - 0.5f;
        sm_point[7] = h + 0.5f;
    }
    __syncthreads();

    // ---- stage 1: mx[t] = sum_p s * motor[k(t,p)] * point[p]  (<= 8 terms)
    const int4* s1 = (const int4*)(stage1 + t * 8);
    const int4 e0 = s1[0], e1 = s1[1];
    {
        const int ee[8] = {e0.x, e0.y, e0.z, e0.w, e1.x, e1.y, e1.z, e1.w};
        float acc = 0.0f;
        #pragma unroll
        for (int p = 0; p < 8; ++p) {
            const int e = ee[p];
            if (e & 0x4000) {
                float v = sm_motor[e & 0xFF] * sm_point[p];
                acc += (e & 0x8000) ? -v : v;
            }
        }
        sm_mx[t] = acc;
    }
    __syncthreads();

    // ---- stage 2: per-thread term for each of 6 outputs (<= 1 term each)
    const int4* s2 = (const int4*)(stage2 + t * 8);
    const int4 f0 = s2[0], f1 = s2[1];
    const float mxv = sm_mx[t];
    float a[6];
    {
        const int ff[6] = {f0.x, f0.y, f0.z, f0.w, f1.x, f1.y};
        #pragma unroll
        for (int r = 0; r < 6; ++r) {
            const int e = ff[r];
            float v = 0.0f;
            if (e & 0x4000) {
                v = mxv * sm_mrev[e & 0xFF];
                if (e & 0x8000) v = -v;
            }
            a[r] = v;
        }
    }
    // wave32 butterfly reduce, then one partial row per wave
    #pragma unroll
    for (int m = 16; m >= 1; m >>= 1) {
        #pragma unroll
        for (int r = 0; r < 6; ++r) a[r] += __shfl_xor(a[r], m, 32);
    }
    if ((t & 31) == 0) {
        const int w = t >> 5;
        #pragma unroll
        for (int r = 0; r < 6; ++r) sm_part[w * 16 + r] = a[r];
    }
    __syncthreads();

#if USE_WMMA_F32
    // Exact fp32 column-sum of the 4x16 partial tile via one WMMA:
    // D = ones(16x4) x B(4x16) + 0  => every D row = colsum(B).
    // B layout (ISA 7.12.2): VGPR0 = rows {0 | lanes0-15, 2 | lanes16-31},
    //                        VGPR1 = rows {1 | lanes0-15, 3 | lanes16-31}.
    const int lane = t & 31;
    const int col  = lane & 15;
    v2f av; av[0] = 1.0f; av[1] = 1.0f;
    v2f bv;
    if (lane < 16) { bv[0] = sm_part[col];      bv[1] = sm_part[16 + col]; }
    else           { bv[0] = sm_part[32 + col]; bv[1] = sm_part[48 + col]; }
    v8f c = {};
    v8f d = __builtin_amdgcn_wmma_f32_16x16x4_f32(false, av, false, bv,
                                                  (short)0, c, false, false);
    // D row M=0 lives in VGPR0 on lanes 0..15 (col = lane)
    if (t < 6) out[(size_t)b * 6 + t] = d[0];
#else
    if (t < 6)
        out[(size_t)b * 6 + t] =
            sm_part[t] + sm_part[16 + t] + sm_part[32 + t] + sm_part[48 + t];
#endif
}

// ---------------------------------------------------------------------------
extern "C" void kernel_launch(void* const* d_in, const int* in_sizes, int n_in,
                              void* d_out, int out_size, void* d_ws, size_t ws_size,
                              hipStream_t stream)
{
    (void)n_in; (void)out_size; (void)ws_size;
    const float* motor = (const float*)d_in[0];
    const float* x     = (const float*)d_in[1];
    const int*   left  = (const int*)d_in[2];
    const int*   right = (const int*)d_in[3];
    const int*   res   = (const int*)d_in[4];
    const int*   sgn   = (const int*)d_in[5];
    const int*   pmask = (const int*)d_in[6];
    const int*   mmask = (const int*)d_in[7];
    const int*   mrs   = (const int*)d_in[8];

    const int nnz    = in_sizes[2];      // 65536
    const int npoint = in_sizes[6];      // 8
    const int nmotor = in_sizes[7];      // 127
    const int nB     = in_sizes[1] / 6;  // 1024

    int* stage1 = (int*)d_ws;
    int* stage2 = stage1 + NODD * 8;

    cga_setup<<<1, 256, 0, stream>>>(left, right, res, sgn, pmask, mmask,
                                     nnz, nmotor, npoint, stage1, stage2);
    cga_apply<<<nB, 128, 0, stream>>>(motor, x, mrs, stage1, stage2,
                                      (float*)d_out, nmotor);
}